// Model_83013127897601
// MI455X (gfx1250) — compile-verified
//
#include <hip/hip_runtime.h>
#include <math.h>

// ---------------------------------------------------------------------------
// Equivariant GNN forward for MI455X (gfx1250, wave32).
// Strategy: fused edge TP -> f32 atomic scatter into kept-column node
// aggregates (fits in 192MB L2), then node linears via V_WMMA_F32_16X16X4_F32.
// Wigner-3j tables computed on-device (exact port of reference math).
// ---------------------------------------------------------------------------

#define N_NODES  30000
#define N_EDGES  120000
#define N_GRAPHS 8

#define AGG0_W 16
#define AGG1_W 736   // kept msg width 728, padded
#define AGG2_W 64

// w3j slot indices (each slot = 256 floats)
#define S101 0
#define S202 1
#define S110 2
#define S111 3
#define S112 4
#define S121 5
#define S122 6
#define S211 7
#define S212 8
#define S220 9
#define S221 10
#define S222 11
#define S312 12
#define S321 13
#define S322 14
#define W3J_FLOATS (15*256)

// workspace layout (in floats)
#define W3J_OFF  ((size_t)0)
#define INV_OFF  ((size_t)4096)
#define SH_OFF   ((size_t)(4096 + 30720))
#define F1_OFF   (SH_OFF  + (size_t)N_EDGES*16)
#define F2_OFF   (F1_OFF  + (size_t)N_NODES*96)
#define F3_OFF   (F2_OFF  + (size_t)N_NODES*160)
#define AGG_OFF  (F3_OFF  + (size_t)N_NODES*8)
#define PRED_OFF (AGG_OFF + (size_t)N_NODES*AGG1_W)

#define SQ3 1.7320508075688772f
#define SQ5 2.23606797749979f
// linear norms: 1/(DENOM*sqrt(Mtot)); shortcut: 1/sqrt(Mtot)
#define NL1_LIN0 0.09622504486493764f   // 1/(1.5*sqrt(48))
#define NL1_SC0  0.17677669529663687f   // 1/sqrt(32)
#define NL1_LIN1 0.16666666666666666f   // 1/(1.5*4)
#define NL1_LIN23 0.08333333333333333f  // 1/(1.5*8)
#define NL1_SC23 0.35355339059327373f   // 1/sqrt(8)
#define NL1_LIN4 0.13608276348795434f   // 1/(1.5*sqrt(24))

typedef float v2f __attribute__((ext_vector_type(2)));
typedef float v8f __attribute__((ext_vector_type(8)));

// ======================= Wigner-3j on device (double) =======================

__device__ double dfac(int n) { double r = 1.0; for (int i = 2; i <= n; ++i) r *= (double)i; return r; }

__device__ double clebsch(int j1,int m1,int j2,int m2,int j3,int m3) {
  if (m3 != m1 + m2) return 0.0;
  int lo = j1 - j2; if (lo < 0) lo = -lo;
  if (j3 < lo || j3 > j1 + j2) return 0.0;
  double pre = sqrt((2.0*j3+1.0)*dfac(j3+j1-j2)*dfac(j3-j1+j2)*dfac(j1+j2-j3)/dfac(j1+j2+j3+1));
  pre *= sqrt(dfac(j3+m3)*dfac(j3-m3)*dfac(j1-m1)*dfac(j1+m1)*dfac(j2-m2)*dfac(j2+m2));
  int kmin = 0;
  if (j2 - j3 - m1 > kmin) kmin = j2 - j3 - m1;
  if (j1 - j3 + m2 > kmin) kmin = j1 - j3 + m2;
  int kmax = j1 + j2 - j3;
  if (j1 - m1 < kmax) kmax = j1 - m1;
  if (j2 + m2 < kmax) kmax = j2 + m2;
  double s = 0.0;
  for (int k = kmin; k <= kmax; ++k) {
    double t = 1.0/(dfac(k)*dfac(j1+j2-j3-k)*dfac(j1-m1-k)*dfac(j2+m2-k)*dfac(j3-j2+m1+k)*dfac(j3-j1-m2+k));
    s += (k & 1) ? -t : t;
  }
  return pre * s;
}

__device__ void umat(int l, double Ur[7][7], double Ui[7][7]) {
  int d = 2*l+1;
  for (int i = 0; i < d; ++i) for (int j = 0; j < d; ++j) { Ur[i][j] = 0.0; Ui[i][j] = 0.0; }
  const double s2 = 0.7071067811865476;
  for (int m = -l; m <= l; ++m) {
    if (m == 0) Ur[l][l] = 1.0;
    else if (m > 0) {
      Ur[l+m][l+m] = ((m & 1) ? -1.0 : 1.0) * s2;
      Ur[l+m][l-m] = s2;
    } else {
      int mu = -m;
      Ui[l+m][l-mu] = s2;                              // 1j*s2
      Ui[l+m][l+mu] = -(((mu & 1) ? -1.0 : 1.0)) * s2; // -1j*(-1)^mu*s2
    }
  }
}

__device__ void w3j_one(int l1, int l2, int l3, float* out) {
  const int d1 = 2*l1+1, d2 = 2*l2+1, d3 = 2*l3+1;
  double C[7][7][7];
  for (int a = 0; a < d1; ++a) for (int b = 0; b < d2; ++b) for (int c = 0; c < d3; ++c) C[a][b][c] = 0.0;
  for (int a = -l1; a <= l1; ++a)
    for (int b = -l2; b <= l2; ++b) {
      int c = a + b;
      if (c >= -l3 && c <= l3) C[l1+a][l2+b][l3+c] = clebsch(l1,a,l2,b,l3,c);
    }
  double U1r[7][7], U1i[7][7], U2r[7][7], U2i[7][7], U3r[7][7], U3i[7][7];
  umat(l1, U1r, U1i); umat(l2, U2r, U2i); umat(l3, U3r, U3i);
  double Kr[7][7][7], Ki[7][7][7];
  double nr = 0.0, ni = 0.0;
  for (int i = 0; i < d1; ++i)
    for (int j = 0; j < d2; ++j)
      for (int k = 0; k < d3; ++k) {
        double sr = 0.0, si = 0.0;
        for (int a = 0; a < d1; ++a) {
          double u1r = U1r[i][a], u1i = -U1i[i][a];
          if (u1r == 0.0 && u1i == 0.0) continue;
          for (int b = 0; b < d2; ++b) {
            double u2r = U2r[j][b], u2i = -U2i[j][b];
            double tr = u1r*u2r - u1i*u2i, ti = u1r*u2i + u1i*u2r;
            if (tr == 0.0 && ti == 0.0) continue;
            for (int c = 0; c < d3; ++c) {
              double cv = C[a][b][c];
              if (cv == 0.0) continue;
              double u3r = U3r[k][c], u3i = U3i[k][c];
              sr += (tr*u3r - ti*u3i)*cv;
              si += (tr*u3i + ti*u3r)*cv;
            }
          }
        }
        Kr[i][j][k] = sr; Ki[i][j][k] = si;
        nr += sr*sr; ni += si*si;
      }
  bool useR = (nr >= ni);
  double nn = sqrt(useR ? nr : ni);
  double T[343];
  int t = 0;
  for (int i = 0; i < d1; ++i)
    for (int j = 0; j < d2; ++j)
      for (int k = 0; k < d3; ++k) T[t++] = (useR ? Kr[i][j][k] : Ki[i][j][k]) / nn;
  int n = d1*d2*d3, bi = 0; double ba = fabs(T[0]);
  for (int q = 1; q < n; ++q) { double aq = fabs(T[q]); if (aq > ba) { ba = aq; bi = q; } }
  double sgn = (T[bi] < 0.0) ? -1.0 : 1.0;
  for (int q = 0; q < n; ++q) out[q] = (float)(T[q]*sgn);
}

__global__ void w3j_kernel(float* __restrict__ w3j) {
  if (threadIdx.x != 0 || blockIdx.x != 0) return;
  const int cmb[15][3] = {{1,0,1},{2,0,2},{1,1,0},{1,1,1},{1,1,2},{1,2,1},{1,2,2},
                          {2,1,1},{2,1,2},{2,2,0},{2,2,1},{2,2,2},{3,1,2},{3,2,1},{3,2,2}};
  for (int s = 0; s < 15; ++s) w3j_one(cmb[s][0], cmb[s][1], cmb[s][2], w3j + s*256);
}

// ======================= Edge geometry: SH + degree =========================

__global__ void edge_sh(const float* __restrict__ pos, const int* __restrict__ snd,
                        const int* __restrict__ rcv, float* __restrict__ sh, float* __restrict__ cnt) {
  int e = blockIdx.x*blockDim.x + threadIdx.x;
  if (e >= N_EDGES) return;
  int s = snd[e], r = rcv[e];
  float rx = pos[r*3+0]-pos[s*3+0], ry = pos[r*3+1]-pos[s*3+1], rz = pos[r*3+2]-pos[s*3+2];
  float rn = sqrtf(rx*rx+ry*ry+rz*rz);
  float inv = 1.f / fmaxf(rn, 1e-9f);
  float x = rx*inv, y = ry*inv, z = rz*inv;
  float* o = sh + (size_t)e*16;
  const float s3 = SQ3, s15 = 3.872983346207417f;
  const float c1 = 0.7905694150420949f, c3 = 0.6123724356957945f;
  o[0]=y; o[1]=z; o[2]=x;
  o[3]=s3*x*y; o[4]=s3*y*z; o[5]=0.5f*(3.f*z*z-1.f); o[6]=s3*x*z; o[7]=0.5f*s3*(x*x-y*y);
  o[8]=c1*y*(3.f*x*x-y*y); o[9]=s15*x*y*z; o[10]=c3*y*(5.f*z*z-1.f);
  o[11]=0.5f*z*(5.f*z*z-3.f); o[12]=c3*x*(5.f*z*z-1.f); o[13]=0.5f*s15*z*(x*x-y*y);
  o[14]=c1*x*(x*x-3.f*y*y); o[15]=0.f;
  atomicAdd(cnt + r, 1.0f);
}

__global__ void invcnt_kernel(float* __restrict__ c) {
  int i = blockIdx.x*blockDim.x + threadIdx.x;
  if (i < N_NODES) c[i] = 1.f / fmaxf(c[i], 1.f);
}

// ======================= Layer 0: edge TP + linear ==========================

__global__ void edge_l0(const float* __restrict__ nf, const float* __restrict__ sh,
                        const float* __restrict__ w3j, const int* __restrict__ snd,
                        const int* __restrict__ rcv, float* __restrict__ agg) {
  int e = blockIdx.x*blockDim.x + threadIdx.x;
  if (e >= N_EDGES) return;
  int s = snd[e], r = rcv[e];
  float f = nf[s];
  const float* sv = sh + (size_t)e*16;
  float* ag = agg + (size_t)r*AGG0_W;
  atomicAdd(ag + 0, f);
  #pragma unroll
  for (int k = 0; k < 3; ++k) {
    float t = 0.f;
    #pragma unroll
    for (int i = 0; i < 3; ++i) t += sv[i]*w3j[S101*256 + i*3 + k];
    atomicAdd(ag + 1 + k, SQ3*f*t);
  }
  #pragma unroll
  for (int k = 0; k < 5; ++k) {
    float t = 0.f;
    #pragma unroll
    for (int i = 0; i < 5; ++i) t += sv[3+i]*w3j[S202*256 + i*5 + k];
    atomicAdd(ag + 4 + k, SQ5*f*t);
  }
}

__global__ void node_linear_l0(const float* __restrict__ agg, const float* __restrict__ invc,
                               const float* __restrict__ Wl0, const float* __restrict__ Wl1,
                               const float* __restrict__ Wl2, const float* __restrict__ Ws0,
                               float* __restrict__ out) {
  int idx = blockIdx.x*blockDim.x + threadIdx.x;
  if (idx >= N_NODES*96) return;
  int n = idx / 96, c = idx - n*96;
  float ic = invc[n];
  const float* a = agg + (size_t)n*AGG0_W;
  float y;
  if (c < 32) {
    y = a[0]*ic * (Wl0[c]*(1.f/1.5f) + Ws0[c]);            // Mtot=1 for both
  } else if (c < 56) {
    int t = c - 32, u = t/3, m = t - u*3;
    y = a[1+m]*ic * Wl1[u] * (1.f/1.5f);
  } else {
    int t = c - 56, u = t/5, m = t - u*5;
    y = a[4+m]*ic * Wl2[u] * (1.f/1.5f);
  }
  out[idx] = y;
}

// ======================= Layer 1: edge TP (kept paths) ======================

template<int D1,int D2,int D3>
__device__ __forceinline__ void mk_sw(const float* __restrict__ shv, const float* __restrict__ w,
                                      float* __restrict__ sw) {
  #pragma unroll
  for (int j = 0; j < D2; ++j)
    #pragma unroll
    for (int k = 0; k < D3; ++k) {
      float s = 0.f;
      #pragma unroll
      for (int i = 0; i < D1; ++i) s += shv[i]*w[(i*D2+j)*D3+k];
      sw[j*D3+k] = s;
    }
}

template<int D2,int D3>
__device__ __forceinline__ void apply8(const float* __restrict__ fv, const float* __restrict__ sw,
                                       float scale, float* __restrict__ dst) {
  #pragma unroll
  for (int u = 0; u < 8; ++u)
    #pragma unroll
    for (int k = 0; k < D3; ++k) {
      float s = 0.f;
      #pragma unroll
      for (int j = 0; j < D2; ++j) s += fv[u*D2+j]*sw[j*D3+k];
      atomicAdd(dst + u*D3 + k, scale*s);
    }
}

template<int D3>
__device__ __forceinline__ void apply_scal32(const float* __restrict__ scal, const float* __restrict__ sw,
                                             float scale, float* __restrict__ dst) {
  for (int u = 0; u < 32; ++u) {
    float f = scal[u]*scale;
    #pragma unroll
    for (int k = 0; k < D3; ++k) atomicAdd(dst + u*D3 + k, f*sw[k]);
  }
}

__global__ __launch_bounds__(128) void edge_l1(const float* __restrict__ feats,
                                               const float* __restrict__ sh,
                                               const float* __restrict__ w3j,
                                               const int* __restrict__ snd,
                                               const int* __restrict__ rcv,
                                               float* __restrict__ agg) {
  __shared__ float s_w[W3J_FLOATS];
  for (int i = threadIdx.x; i < W3J_FLOATS; i += blockDim.x) s_w[i] = w3j[i];
  __syncthreads();

  int e = blockIdx.x*blockDim.x + threadIdx.x;
  if (e >= N_EDGES) return;
  int s = snd[e], r = rcv[e];
  const float* fp = feats + (size_t)s*96;
  __builtin_prefetch(fp, 0, 3);
  const float* sv = sh + (size_t)e*16;
  float sh1[3], sh2[5], sh3[7];
  #pragma unroll
  for (int i = 0; i < 3; ++i) sh1[i] = sv[i];
  #pragma unroll
  for (int i = 0; i < 5; ++i) sh2[i] = sv[3+i];
  #pragma unroll
  for (int i = 0; i < 7; ++i) sh3[i] = sv[8+i];
  float v1[24], v2[40];
  #pragma unroll
  for (int i = 0; i < 24; ++i) v1[i] = fp[32+i];
  #pragma unroll
  for (int i = 0; i < 40; ++i) v2[i] = fp[56+i];
  float* ag = agg + (size_t)r*AGG1_W;

  // fb block (I0 -> G0[0..31], I1 -> G2 ch0..7, I2 -> G3 ch0..7)
  for (int c = 0; c < 32; ++c) atomicAdd(ag + c, fp[c]);
  #pragma unroll
  for (int t = 0; t < 24; ++t) atomicAdd(ag + 96 + t, v1[t]);
  #pragma unroll
  for (int t = 0; t < 40; ++t) atomicAdd(ag + 288 + t, v2[t]);

  float sw[25];
  mk_sw<3,1,3>(sh1, s_w + S101*256, sw);  apply_scal32<3>(fp, sw, SQ3, ag + 96 + 8*3);    // P0
  mk_sw<3,3,1>(sh1, s_w + S110*256, sw);  apply8<3,1>(v1, sw, 1.f, ag + 32);              // P1
  mk_sw<3,3,3>(sh1, s_w + S111*256, sw);  apply8<3,3>(v1, sw, SQ3, ag + 48);              // P2
  mk_sw<3,3,5>(sh1, s_w + S112*256, sw);  apply8<3,5>(v1, sw, SQ5, ag + 288 + 8*5);       // P3
  mk_sw<3,5,3>(sh1, s_w + S121*256, sw);  apply8<5,3>(v2, sw, SQ3, ag + 96 + 40*3);       // P4
  mk_sw<3,5,5>(sh1, s_w + S122*256, sw);  apply8<5,5>(v2, sw, SQ5, ag + 608);             // P5
  mk_sw<5,1,5>(sh2, s_w + S202*256, sw);  apply_scal32<5>(fp, sw, SQ5, ag + 288 + 16*5);  // P7
  mk_sw<5,3,3>(sh2, s_w + S211*256, sw);  apply8<3,3>(v1, sw, SQ3, ag + 96 + 48*3);       // P8
  mk_sw<5,3,5>(sh2, s_w + S212*256, sw);  apply8<3,5>(v1, sw, SQ5, ag + 608 + 8*5);       // P9
  mk_sw<5,5,1>(sh2, s_w + S220*256, sw);  apply8<5,1>(v2, sw, 1.f, ag + 40);              // P11
  mk_sw<5,5,3>(sh2, s_w + S221*256, sw);  apply8<5,3>(v2, sw, SQ3, ag + 48 + 8*3);        // P12
  mk_sw<5,5,5>(sh2, s_w + S222*256, sw);  apply8<5,5>(v2, sw, SQ5, ag + 288 + 48*5);      // P13
  mk_sw<7,3,5>(sh3, s_w + S312*256, sw);  apply8<3,5>(v1, sw, SQ5, ag + 288 + 56*5);      // P17
  mk_sw<7,5,3>(sh3, s_w + S321*256, sw);  apply8<5,3>(v2, sw, SQ3, ag + 96 + 56*3);       // P20
  mk_sw<7,5,5>(sh3, s_w + S322*256, sw);  apply8<5,5>(v2, sw, SQ5, ag + 608 + 16*5);      // P21
}

// ======================= Layer 1: node linear via WMMA ======================
// A(16x4 f32): lanes0-15 K={0,1}, lanes16-31 K={2,3}.  B(4x16): same K split,
// N = lane&15.  C/D v8f: VGPR r -> row r (lanes0-15) / r+8 (lanes16-31).

__device__ __forceinline__ v8f gemm_step(const float* __restrict__ arow, float ic, int lane,
                                         int colBase, int st, int m, int nch,
                                         const float* __restrict__ W, int mo, int ncolBase,
                                         float norm, v8f c) {
  const int kb  = (lane >> 4) << 1;
  const int nc0 = ncolBase + (lane & 15);
  const float wn = (nc0 < mo) ? norm : 0.f;
  const int nc   = (nc0 < mo) ? nc0 : 0;
  for (int k0 = 0; k0 < nch; k0 += 4) {
    int u = k0 + kb;
    v2f a, b;
    a.x = arow[colBase + u*st + m] * ic;
    a.y = arow[colBase + (u+1)*st + m] * ic;
    b.x = W[u*mo + nc] * wn;
    b.y = W[(u+1)*mo + nc] * wn;
    c = __builtin_amdgcn_wmma_f32_16x16x4_f32(false, a, false, b, (short)0, c, false, false);
  }
  return c;
}

__device__ __forceinline__ void store_tile(float* __restrict__ out, int tb, int lane,
                                           int outBase, int st, int m, int mo, int ncolBase, v8f c) {
  int nc = ncolBase + (lane & 15);
  if (nc >= mo) return;
  int half = (lane >> 4) * 8;
  #pragma unroll
  for (int r2 = 0; r2 < 8; ++r2) {
    int node = tb + half + r2;
    if (node < N_NODES) out[(size_t)node*160 + outBase + nc*st + m] = c[r2];
  }
}

__global__ __launch_bounds__(256) void node_linear_l1(
    const float* __restrict__ agg, const float* __restrict__ invc,
    const float* __restrict__ Wl0, const float* __restrict__ Wl1, const float* __restrict__ Wl2,
    const float* __restrict__ Wl3, const float* __restrict__ Wl4,
    const float* __restrict__ Ws0, const float* __restrict__ Ws2, const float* __restrict__ Ws3,
    float* __restrict__ out) {
  const int lane = threadIdx.x & 31;
  const int tile = blockIdx.x*8 + (threadIdx.x >> 5);
  const int tb   = tile*16;
  const int ra   = tb + (lane & 15);
  const float ic = (ra < N_NODES) ? invc[ra] : 0.f;
  const float* arow = agg + (size_t)(ra < N_NODES ? ra : 0)*AGG1_W;

  // out0: (32,0,1). lin K=48 (G0), sc K=32 (fb block = G0 ch0..31)
  for (int nt = 0; nt < 2; ++nt) {
    v8f c = {};
    c = gemm_step(arow, ic, lane, 0, 1, 0, 48, Wl0, 32, nt*16, NL1_LIN0, c);
    c = gemm_step(arow, ic, lane, 0, 1, 0, 32, Ws0, 32, nt*16, NL1_SC0,  c);
    store_tile(out, tb, lane, 0, 1, 0, 32, nt*16, c);
  }
  // out1: (8,1,1). lin K=16 (G1)
  for (int m = 0; m < 3; ++m) {
    v8f c = {};
    c = gemm_step(arow, ic, lane, 48, 3, m, 16, Wl1, 8, 0, NL1_LIN1, c);
    store_tile(out, tb, lane, 32, 3, m, 8, 0, c);
  }
  // out2: (8,1,-1). lin K=64 (G2), sc K=8 (G2 ch0..7)
  for (int m = 0; m < 3; ++m) {
    v8f c = {};
    c = gemm_step(arow, ic, lane, 96, 3, m, 64, Wl2, 8, 0, NL1_LIN23, c);
    c = gemm_step(arow, ic, lane, 96, 3, m, 8,  Ws2, 8, 0, NL1_SC23,  c);
    store_tile(out, tb, lane, 56, 3, m, 8, 0, c);
  }
  // out3: (8,2,1). lin K=64 (G3), sc K=8 (G3 ch0..7)
  for (int m = 0; m < 5; ++m) {
    v8f c = {};
    c = gemm_step(arow, ic, lane, 288, 5, m, 64, Wl3, 8, 0, NL1_LIN23, c);
    c = gemm_step(arow, ic, lane, 288, 5, m, 8,  Ws3, 8, 0, NL1_SC23,  c);
    store_tile(out, tb, lane, 80, 5, m, 8, 0, c);
  }
  // out4: (8,2,-1). lin K=24 (G4)
  for (int m = 0; m < 5; ++m) {
    v8f c = {};
    c = gemm_step(arow, ic, lane, 608, 5, m, 24, Wl4, 8, 0, NL1_LIN4, c);
    store_tile(out, tb, lane, 120, 5, m, 8, 0, c);
  }
}

// ======================= Layer 2: edge TP + linear ==========================

__global__ void edge_l2(const float* __restrict__ feats, const float* __restrict__ sh,
                        const float* __restrict__ w3j, const int* __restrict__ snd,
                        const int* __restrict__ rcv, float* __restrict__ agg) {
  int e = blockIdx.x*blockDim.x + threadIdx.x;
  if (e >= N_EDGES) return;
  int s = snd[e], r = rcv[e];
  const float* fp = feats + (size_t)s*160;
  __builtin_prefetch(fp, 0, 3);
  const float* sv = sh + (size_t)e*16;
  float* ag = agg + (size_t)r*AGG2_W;

  for (int c = 0; c < 32; ++c) atomicAdd(ag + c, fp[c]);   // fb I0

  float sw1[3], sw2[5];
  #pragma unroll
  for (int j = 0; j < 3; ++j) {
    float t = 0.f;
    #pragma unroll
    for (int i = 0; i < 3; ++i) t += sv[i]*w3j[S110*256 + i*3 + j];
    sw1[j] = t;
  }
  #pragma unroll
  for (int j = 0; j < 5; ++j) {
    float t = 0.f;
    #pragma unroll
    for (int i = 0; i < 5; ++i) t += sv[3+i]*w3j[S220*256 + i*5 + j];
    sw2[j] = t;
  }
  #pragma unroll
  for (int u = 0; u < 8; ++u) {
    float pa = 0.f, pb = 0.f;
    #pragma unroll
    for (int j = 0; j < 3; ++j) { pa += fp[32+u*3+j]*sw1[j]; pb += fp[56+u*3+j]*sw1[j]; }
    atomicAdd(ag + 48 + u, pa);   // (8,0,-1) from v1(+)
    atomicAdd(ag + 32 + u, pb);   // (8,0,+1) from v1(-)
    float pc = 0.f, pd = 0.f;
    #pragma unroll
    for (int j = 0; j < 5; ++j) { pc += fp[80+u*5+j]*sw2[j]; pd += fp[120+u*5+j]*sw2[j]; }
    atomicAdd(ag + 40 + u, pc);   // (8,0,+1) from v2(+)
    atomicAdd(ag + 56 + u, pd);   // (8,0,-1) from v2(-)
  }
}

__global__ void node_linear_l2(const float* __restrict__ agg, const float* __restrict__ invc,
                               const float* __restrict__ W20, const float* __restrict__ W21,
                               const float* __restrict__ Ws1, float* __restrict__ f3) {
  int idx = blockIdx.x*blockDim.x + threadIdx.x;
  if (idx >= N_NODES*8) return;
  int n = idx >> 3, c = idx & 7;
  const float* a = agg + (size_t)n*AGG2_W;
  float ic = invc[n];
  float y;
  if (c == 0) {                       // (1,0,-1): K=16 at cols 48..63
    float s = 0.f;
    #pragma unroll
    for (int u = 0; u < 16; ++u) s += a[48+u]*W20[u];
    y = ic*s*(1.f/6.f);               // 1/(1.5*sqrt(16))
  } else {                            // (7,0,1): lin K=48 cols0..47, sc K=32 cols0..31
    int o = c - 1;
    float s48 = 0.f, s32 = 0.f;
    for (int u = 0; u < 48; ++u) s48 += a[u]*W21[u*7+o];
    for (int u = 0; u < 32; ++u) s32 += a[u]*Ws1[u*7+o];
    y = ic*(s48*0.09622504486493764f + s32*0.17677669529663687f);
  }
  f3[idx] = y;
}

// ======================= Readout =============================================

__global__ void pred_accum(const float* __restrict__ f3, const int* __restrict__ batch,
                           float* __restrict__ pred) {
  int idx = blockIdx.x*blockDim.x + threadIdx.x;
  if (idx >= N_NODES*8) return;
  int n = idx >> 3, c = idx & 7;
  atomicAdd(pred + batch[n]*8 + c, f3[idx]);
}

__global__ void combine(const float* __restrict__ pred, float* __restrict__ out) {
  int g = threadIdx.x;
  if (g >= N_GRAPHS) return;
  float odd = pred[g*8+0], e1 = pred[g*8+1];
  out[g*8+0] = odd*e1;
  out[g*8+1] = -odd*e1;
  #pragma unroll
  for (int j = 0; j < 6; ++j) out[g*8+2+j] = pred[g*8+2+j];
}

// ======================= Host launcher ======================================

extern "C" void kernel_launch(void* const* d_in, const int* in_sizes, int n_in,
                              void* d_out, int out_size, void* d_ws, size_t ws_size,
                              hipStream_t stream) {
  const float* node_features = (const float*)d_in[0];
  const float* pos           = (const float*)d_in[1];
  // params leaves in sorted pytree order
  const float* Wl0_lin0 = (const float*)d_in[2];   // (1,32)
  const float* Wl0_lin1 = (const float*)d_in[3];   // (1,8)
  const float* Wl0_lin2 = (const float*)d_in[4];   // (1,8)
  const float* Wl0_sc0  = (const float*)d_in[5];   // (1,32)
  const float* Wl1_lin0 = (const float*)d_in[6];   // (48,32)
  const float* Wl1_lin1 = (const float*)d_in[7];   // (16,8)
  const float* Wl1_lin2 = (const float*)d_in[8];   // (64,8)
  const float* Wl1_lin3 = (const float*)d_in[9];   // (64,8)
  const float* Wl1_lin4 = (const float*)d_in[10];  // (24,8)
  const float* Wl1_sc0  = (const float*)d_in[11];  // (32,32)
  const float* Wl1_sc2  = (const float*)d_in[12];  // (8,8)
  const float* Wl1_sc3  = (const float*)d_in[13];  // (8,8)
  const float* Wl2_lin0 = (const float*)d_in[14];  // (16,1)
  const float* Wl2_lin1 = (const float*)d_in[15];  // (48,7)
  const float* Wl2_sc1  = (const float*)d_in[16];  // (32,7)
  const int* senders   = (const int*)d_in[17];
  const int* receivers = (const int*)d_in[18];
  const int* batch     = (const int*)d_in[19];
  float* out = (float*)d_out;

  float* ws   = (float*)d_ws;
  float* w3j  = ws + W3J_OFF;
  float* invc = ws + INV_OFF;
  float* sh   = ws + SH_OFF;
  float* f1   = ws + F1_OFF;
  float* f2   = ws + F2_OFF;
  float* f3   = ws + F3_OFF;
  float* agg  = ws + AGG_OFF;
  float* pred = ws + PRED_OFF;

  // 1) Wigner-3j tables + geometry
  w3j_kernel<<<1, 32, 0, stream>>>(w3j);
  hipMemsetAsync(invc, 0, (size_t)N_NODES*sizeof(float), stream);
  edge_sh<<<(N_EDGES+255)/256, 256, 0, stream>>>(pos, senders, receivers, sh, invc);
  invcnt_kernel<<<(N_NODES+255)/256, 256, 0, stream>>>(invc);

  // 2) Layer 0
  hipMemsetAsync(agg, 0, (size_t)N_NODES*AGG0_W*sizeof(float), stream);
  edge_l0<<<(N_EDGES+255)/256, 256, 0, stream>>>(node_features, sh, w3j, senders, receivers, agg);
  node_linear_l0<<<(N_NODES*96+255)/256, 256, 0, stream>>>(agg, invc, Wl0_lin0, Wl0_lin1, Wl0_lin2, Wl0_sc0, f1);

  // 3) Layer 1 (fused TP scatter + WMMA node linear)
  hipMemsetAsync(agg, 0, (size_t)N_NODES*AGG1_W*sizeof(float), stream);
  edge_l1<<<(N_EDGES+127)/128, 128, 0, stream>>>(f1, sh, w3j, senders, receivers, agg);
  {
    int tiles = (N_NODES + 15)/16;
    int blocks = (tiles + 7)/8;
    node_linear_l1<<<blocks, 256, 0, stream>>>(agg, invc,
        Wl1_lin0, Wl1_lin1, Wl1_lin2, Wl1_lin3, Wl1_lin4, Wl1_sc0, Wl1_sc2, Wl1_sc3, f2);
  }

  // 4) Layer 2
  hipMemsetAsync(agg, 0, (size_t)N_NODES*AGG2_W*sizeof(float), stream);
  edge_l2<<<(N_EDGES+255)/256, 256, 0, stream>>>(f2, sh, w3j, senders, receivers, agg);
  node_linear_l2<<<(N_NODES*8+255)/256, 256, 0, stream>>>(agg, invc, Wl2_lin0, Wl2_lin1, Wl2_sc1, f3);

  // 5) Readout
  hipMemsetAsync(pred, 0, (size_t)N_GRAPHS*8*sizeof(float), stream);
  pred_accum<<<(N_NODES*8+255)/256, 256, 0, stream>>>(f3, batch, pred);
  combine<<<1, 32, 0, stream>>>(pred, out);

  (void)in_sizes; (void)n_in; (void)out_size; (void)ws_size;
}